// GNN_16003048145622
// MI455X (gfx1250) — compile-verified
//
#include <hip/hip_runtime.h>
#include <math.h>

// ---------------- problem constants ----------------
#define NN   100000          // nodes
#define EE   1600000         // edges
#define GG   64              // graphs
#define NT   (NN / 16)       // 6250 16-row tiles (exact)
#define BNEPS 1e-5f
#define HP   34              // padded row stride for head staging (bank-conflict-free)

typedef __attribute__((ext_vector_type(2))) float v2f;
typedef __attribute__((ext_vector_type(8))) float v8f;

// float atomic max via int ordering trick (works for mixed signs, -inf init)
__device__ __forceinline__ void atomic_max_f32(float* addr, float val) {
    if (val >= 0.0f) atomicMax((int*)addr, __float_as_int(val));
    else             atomicMin((unsigned int*)addr, __float_as_uint(val));
}

__device__ __forceinline__ float elu1(float v) {
    return v > 0.0f ? v : (expf(v) - 1.0f);
}

// ---------------- init / small kernels ----------------
__global__ void init_out_kernel(float* __restrict__ out, float* __restrict__ maxbuf) {
    int i = blockIdx.x * blockDim.x + threadIdx.x;
    if (i < GG * 16) { out[i] = 0.0f; maxbuf[i] = -__builtin_huge_valf(); }
}

__global__ void addmax_kernel(float* __restrict__ out, float* __restrict__ maxbuf) {
    int i = blockIdx.x * blockDim.x + threadIdx.x;
    if (i < GG * 16) { out[i] += maxbuf[i]; maxbuf[i] = -__builtin_huge_valf(); }
}

__global__ void zero4_kernel(float4* __restrict__ p, int n4) {
    int i = blockIdx.x * blockDim.x + threadIdx.x;
    if (i < n4) p[i] = make_float4(0.f, 0.f, 0.f, 0.f);
}

// ---------------- edge scatter-add: agg[dst] += x[src] ----------------
// 16 threads per edge, each handles 4 consecutive features (float4 gather).
// x/agg are L2-resident (25.6 MB each); atomics resolve at L2.
__global__ void edge_kernel(const float* __restrict__ x,
                            const long long* __restrict__ ei,
                            float* __restrict__ agg) {
    long long gid = (long long)blockIdx.x * blockDim.x + threadIdx.x;
    if (gid >= (long long)EE * 16) return;
    int e = (int)(gid >> 4);
    int q = ((int)gid & 15) << 2;
    long long s = ei[e];
    long long d = ei[(long long)EE + e];
    float4 v = *(const float4*)(x + s * 64 + q);
    float* p = agg + d * 64 + q;
    atomicAdd(p + 0, v.x);
    atomicAdd(p + 1, v.y);
    atomicAdd(p + 2, v.z);
    atomicAdd(p + 3, v.w);
}

// Stage a (din x dout) row-major weight into WMMA-B-operand layout:
// pair index ((kk4*ntiles + j)*32 + L) holds {W[k0][c], W[k0+1][c]},
// k0 = kk4*4 + 2*(L>>4), c = j*16 + (L&15). Each lane then reads one
// contiguous, conflict-free ds_load_b64 per WMMA.
__device__ __forceinline__ void stage_bops(const float* __restrict__ W,
                                           float* __restrict__ sWb,
                                           int din, int dout, int ntiles,
                                           int tid, int nthreads) {
    int npairs = (din / 4) * ntiles * 32;
    for (int idx = tid; idx < npairs; idx += nthreads) {
        int L   = idx & 31;
        int j   = (idx >> 5) % ntiles;
        int kk4 = idx / (32 * ntiles);
        int k0  = kk4 * 4 + ((L >> 4) << 1);
        int c   = j * 16 + (L & 15);
        sWb[idx * 2 + 0] = W[k0 * dout + c];
        sWb[idx * 2 + 1] = W[(k0 + 1) * dout + c];
    }
}

// ---------------- GINConv MLP: xout = ELU(BN((x+agg)@W + b)) ----------------
// One wave per 16-node tile; 4x v_wmma_f32_16x16x4_f32 tiles across 64 cols.
__global__ __launch_bounds__(256) void conv_kernel(
    const float* __restrict__ x, const float* __restrict__ agg,
    const float* __restrict__ W, const float* __restrict__ bias,
    const float* __restrict__ bnb, const float* __restrict__ bng,
    const float* __restrict__ bnrm, const float* __restrict__ bnrv,
    float* __restrict__ xout) {
    __shared__ float sWb[64 * 64];       // B-operand-swizzled weights
    int tid = threadIdx.x;
    stage_bops(W, sWb, 64, 64, 4, tid, 256);
    __syncthreads();

    int wave = tid >> 5, lane = tid & 31;
    int tile = blockIdx.x * 8 + wave;
    if (tile >= NT) return;
    int row = lane & 15;
    int kb  = (lane >> 4) << 1;          // lanes 16-31 hold K+2,K+3
    int node = tile * 16 + row;
    const float* xr = x + (long long)node * 64;
    const float* ar = agg + (long long)node * 64;

    v8f c[4] = {{}, {}, {}, {}};
    for (int kk = 0; kk < 64; kk += 4) {
        v2f xa = *(const v2f*)(xr + kk + kb);
        v2f aa = *(const v2f*)(ar + kk + kb);
        v2f a  = xa + aa;                 // v_pk_add_f32
        const float* bp = &sWb[((kk >> 2) * 4) * 64 + lane * 2];
#pragma unroll
        for (int j = 0; j < 4; ++j) {
            v2f b = *(const v2f*)(bp + j * 64);
            c[j] = __builtin_amdgcn_wmma_f32_16x16x4_f32(
                false, a, false, b, (short)0, c[j], false, false);
        }
    }
#pragma unroll
    for (int j = 0; j < 4; ++j) {
        int col = j * 16 + row;
        float s = bng[col] * rsqrtf(bnrv[col] + BNEPS);
        float o = bias[col] - bnrm[col];
        float t = bnb[col];
#pragma unroll
        for (int r = 0; r < 8; ++r) {
            int M = r + ((lane >> 4) << 3);
            float v = (c[j][r] + o) * s + t;
            xout[(long long)(tile * 16 + M) * 64 + col] = elu1(v);
        }
    }
}

// ---------------- score head: z = BN2(ELU(BN1(x@W1+b1))@W2+b2) ----------------
// Z (N x 16) written/accumulated; per-graph atomic segment-max into maxbuf.
__global__ __launch_bounds__(256) void head_kernel(
    const float* __restrict__ x, const long long* __restrict__ batch,
    const float* __restrict__ W1, const float* __restrict__ W2,
    const float* __restrict__ b1, const float* __restrict__ b2,
    const float* __restrict__ bn1b, const float* __restrict__ bn1g,
    const float* __restrict__ bn1rm, const float* __restrict__ bn1rv,
    const float* __restrict__ bn2b, const float* __restrict__ bn2g,
    const float* __restrict__ bn2rm, const float* __restrict__ bn2rv,
    float* __restrict__ Z, float* __restrict__ maxbuf, int accumulate) {
    __shared__ float sW1b[16 * 2 * 64];  // 64x32 weights, B-op layout
    __shared__ float sW2b[8 * 1 * 64];   // 32x16 weights, B-op layout
    __shared__ float sH[8][16 * HP];     // per-wave C->A relayout staging (padded)
    int tid = threadIdx.x;
    stage_bops(W1, sW1b, 64, 32, 2, tid, 256);
    stage_bops(W2, sW2b, 32, 16, 1, tid, 256);
    __syncthreads();

    int wave = tid >> 5, lane = tid & 31;
    int tile = blockIdx.x * 8 + wave;
    if (tile >= NT) return;
    int row = lane & 15;
    int kb  = (lane >> 4) << 1;
    int node = tile * 16 + row;
    const float* xr = x + (long long)node * 64;

    // GEMM1: 16x64 @ 64x32 -> two 16x16 fp32 accumulators
    v8f c0 = {}, c1 = {};
    for (int kk = 0; kk < 64; kk += 4) {
        v2f a = *(const v2f*)(xr + kk + kb);
        const float* bp = &sW1b[((kk >> 2) * 2) * 64 + lane * 2];
        v2f bA = *(const v2f*)(bp);
        v2f bB = *(const v2f*)(bp + 64);
        c0 = __builtin_amdgcn_wmma_f32_16x16x4_f32(false, a, false, bA, (short)0, c0, false, false);
        c1 = __builtin_amdgcn_wmma_f32_16x16x4_f32(false, a, false, bB, (short)0, c1, false, false);
    }

    // bias + BN1 + ELU, stage 16x32 (stride HP) into per-wave LDS
    {
        int cA = row, cB = 16 + row;
        float sA = bn1g[cA] * rsqrtf(bn1rv[cA] + BNEPS);
        float oA = b1[cA] - bn1rm[cA];
        float tA = bn1b[cA];
        float sB = bn1g[cB] * rsqrtf(bn1rv[cB] + BNEPS);
        float oB = b1[cB] - bn1rm[cB];
        float tB = bn1b[cB];
        float* Hs = &sH[wave][0];
#pragma unroll
        for (int r = 0; r < 8; ++r) {
            int M = r + ((lane >> 4) << 3);
            Hs[M * HP + cA] = elu1((c0[r] + oA) * sA + tA);
            Hs[M * HP + cB] = elu1((c1[r] + oB) * sB + tB);
        }
    }
    // LDS ops from the same wave are in-order: RAW through sH is safe.

    // GEMM2: 16x32 @ 32x16 -> one 16x16 tile (A reads conflict-free, b64)
    v8f c2 = {};
    {
        const float* Hs = &sH[wave][0];
        for (int kk = 0; kk < 32; kk += 4) {
            v2f a = *(const v2f*)(Hs + row * HP + kk + kb);
            v2f b = *(const v2f*)(&sW2b[(kk >> 2) * 64 + lane * 2]);
            c2 = __builtin_amdgcn_wmma_f32_16x16x4_f32(false, a, false, b, (short)0, c2, false, false);
        }
    }

    // bias + BN2; write/accumulate Z; atomic per-graph max of this layer's z
    {
        int col = row;
        float s2 = bn2g[col] * rsqrtf(bn2rv[col] + BNEPS);
        float o2 = b2[col] - bn2rm[col];
        float t2 = bn2b[col];
#pragma unroll
        for (int r = 0; r < 8; ++r) {
            int M = r + ((lane >> 4) << 3);
            int nd = tile * 16 + M;
            float z = (c2[r] + o2) * s2 + t2;
            float za = accumulate ? (Z[(long long)nd * 16 + col] + z) : z;
            Z[(long long)nd * 16 + col] = za;
            long long g = batch[nd];
            atomic_max_f32(&maxbuf[g * 16 + col], z);
        }
    }
}

// ---------------- host launcher ----------------
extern "C" void kernel_launch(void* const* d_in, const int* in_sizes, int n_in,
                              void* d_out, int out_size, void* d_ws, size_t ws_size,
                              hipStream_t stream) {
    (void)n_in; (void)out_size; (void)ws_size;
    // Input flattening: prefer insertion order (x, edge_index, batch, params);
    // fall back to fully key-sorted flattening (batch, edge_index, params..., x).
    int ix, iei, ib, pb;
    if (in_sizes[0] == NN * 64) { ix = 0; iei = 1; ib = 2; pb = 3; }
    else                        { ib = 0; iei = 1; pb = 2; ix = pb + 48; }

    const float*     x0    = (const float*)d_in[ix];
    const long long* ei    = (const long long*)d_in[iei];
    const long long* batch = (const long long*)d_in[ib];

    // params (JAX sorted-key order): conv[0..1]{W,b,bn.b,bn.g,bn.rm,bn.rv},
    // then lin[0..2]{W1,W2,b1,b2,bn1.{b,g,rm,rv},bn2.{b,g,rm,rv}}
    const float* cv[2][6];
    for (int l = 0; l < 2; ++l)
        for (int k = 0; k < 6; ++k) cv[l][k] = (const float*)d_in[pb + l * 6 + k];
    const float* ln[3][12];
    for (int l = 0; l < 3; ++l)
        for (int k = 0; k < 12; ++k) ln[l][k] = (const float*)d_in[pb + 12 + l * 12 + k];

    float* out = (float*)d_out;                 // 64 x 16
    float* Z   = out + GG * 16;                 // N x 16
    float* X   = Z + (long long)NN * 16;        // N x 64 (intermediate + final x)
    float* agg    = (float*)d_ws;               // N x 64 scratch (25.6 MB)
    float* maxbuf = agg + (long long)NN * 64;   // 64 x 16 per-layer segment-max

    const int TB = 256;
    const int gHC  = (NT + 7) / 8;              // 782 blocks for head/conv
    const int gZ   = (NN * 64 / 4 + TB - 1) / TB;
    const int gE   = (int)(((long long)EE * 16 + TB - 1) / TB);

    init_out_kernel<<<4, TB, 0, stream>>>(out, maxbuf);

    // layer 0 head on raw x
    head_kernel<<<gHC, TB, 0, stream>>>(x0, batch, ln[0][0], ln[0][1], ln[0][2], ln[0][3],
                                        ln[0][4], ln[0][5], ln[0][6], ln[0][7],
                                        ln[0][8], ln[0][9], ln[0][10], ln[0][11],
                                        Z, maxbuf, 0);
    addmax_kernel<<<4, TB, 0, stream>>>(out, maxbuf);

    // layer 1: aggregate on x0, conv -> X, head on X
    zero4_kernel<<<gZ, TB, 0, stream>>>((float4*)agg, NN * 64 / 4);
    edge_kernel<<<gE, TB, 0, stream>>>(x0, ei, agg);
    conv_kernel<<<gHC, TB, 0, stream>>>(x0, agg, cv[0][0], cv[0][1], cv[0][2], cv[0][3],
                                        cv[0][4], cv[0][5], X);
    head_kernel<<<gHC, TB, 0, stream>>>(X, batch, ln[1][0], ln[1][1], ln[1][2], ln[1][3],
                                        ln[1][4], ln[1][5], ln[1][6], ln[1][7],
                                        ln[1][8], ln[1][9], ln[1][10], ln[1][11],
                                        Z, maxbuf, 1);
    addmax_kernel<<<4, TB, 0, stream>>>(out, maxbuf);

    // layer 2: aggregate on X, conv in-place (row-wise independent), head on X
    zero4_kernel<<<gZ, TB, 0, stream>>>((float4*)agg, NN * 64 / 4);
    edge_kernel<<<gE, TB, 0, stream>>>(X, ei, agg);
    conv_kernel<<<gHC, TB, 0, stream>>>(X, agg, cv[1][0], cv[1][1], cv[1][2], cv[1][3],
                                        cv[1][4], cv[1][5], X);
    head_kernel<<<gHC, TB, 0, stream>>>(X, batch, ln[2][0], ln[2][1], ln[2][2], ln[2][3],
                                        ln[2][4], ln[2][5], ln[2][6], ln[2][7],
                                        ln[2][8], ln[2][9], ln[2][10], ln[2][11],
                                        Z, maxbuf, 1);
    addmax_kernel<<<4, TB, 0, stream>>>(out, maxbuf);
}